// MultiHeadAttention_22402549416392
// MI455X (gfx1250) — compile-verified
//
#include <hip/hip_runtime.h>
#include <hip/hip_bf16.h>

typedef __attribute__((ext_vector_type(2))) float v2f;
typedef __attribute__((ext_vector_type(8))) float v8f;

__device__ __forceinline__ v8f wmma4(v2f a, v2f b, v8f c) {
    // V_WMMA_F32_16X16X4_F32: D = A(16x4) x B(4x16) + C(16x16), all fp32
    return __builtin_amdgcn_wmma_f32_16x16x4_f32(
        false, a, false, b, (short)0, c, false, false);
}

// CDNA5 async global->LDS DMA (ASYNCcnt path, CDNA5 ISA).
// GVS mode: mem_addr = SADDR(base) + VADDR(per-lane byte offset); VDST holds
// the per-lane LDS byte address (low 32 bits of the flat shared pointer).
__device__ __forceinline__ void async_lds_b64(unsigned lds_byte,
                                              const float* base,
                                              unsigned gbyte) {
    asm volatile("global_load_async_to_lds_b64 %0, %1, %2"
                 :: "v"(lds_byte), "v"(gbyte), "s"(base) : "memory");
}
__device__ __forceinline__ void async_lds_b32(unsigned lds_byte,
                                              const float* base,
                                              unsigned gbyte) {
    asm volatile("global_load_async_to_lds_b32 %0, %1, %2"
                 :: "v"(lds_byte), "v"(gbyte), "s"(base) : "memory");
}
__device__ __forceinline__ void wait_async_6() {
    asm volatile("s_wait_asynccnt 0x6" ::: "memory");
}
__device__ __forceinline__ void wait_async_0() {
    asm volatile("s_wait_asynccnt 0x0" ::: "memory");
}
__device__ __forceinline__ unsigned lds_off(const float* p) {
    return (unsigned)(uintptr_t)p;   // low 32 bits of flat LDS addr = LDS offset
}

// ---------------------------------------------------------------------------
// GEMM: C[m][n] = sum_k A[m][k] * W[n][k] + bias[n]   (torch Linear: x @ W^T + b)
// M = 8192, K = 768, N = 768. Block tile 64(m) x 128(n), 256 threads = 8 waves.
// Wave w: m-tile (w&3), n-tiles (w>>2)*4 .. +3  -> 4 accumulators of 16x16.
// LDS double-buffered; tiles staged with global_load_async_to_lds_b64 and the
// next K-chunk prefetched while the current one feeds 16 WMMAs per wave.
// ---------------------------------------------------------------------------
__global__ __launch_bounds__(256) void gemm_xwT_bias(
    const float* __restrict__ A, const float* __restrict__ W,
    const float* __restrict__ bias, float* __restrict__ C) {
    const int K = 768, N = 768;
    // stride 18: even (8B-aligned B64 targets), 18*r mod 64 distinct for r<16
    __shared__ float asb[2][64 * 18];
    __shared__ float wsb[2][128 * 18];

    const int tid  = threadIdx.x;
    const int lane = tid & 31;
    const int wave = tid >> 5;
    const int mBase = blockIdx.x * 64;
    const int nBase = blockIdx.y * 128;
    const int mt = wave & 3;
    const int ng = (wave >> 2) * 4;

    v8f acc[4];
    const v8f vzero = {0.f,0.f,0.f,0.f,0.f,0.f,0.f,0.f};
#pragma unroll
    for (int t = 0; t < 4; ++t) acc[t] = vzero;

    // B64 staging: pair-index tid -> row = tid>>3 (0..31), colpair = (tid&7)*2
    const int prow = tid >> 3;
    const int pc   = (tid & 7) * 2;

    auto issue = [&](int k0, int b) {
        // A tile 64x16: rows prow, prow+32  (2 full-EXEC issues)
#pragma unroll
        for (int j = 0; j < 2; ++j) {
            const int rr = prow + j * 32;
            async_lds_b64(lds_off(&asb[b][rr * 18 + pc]), A,
                          (unsigned)(((mBase + rr) * K + k0 + pc) * 4));
        }
        // W tile 128x16: rows prow, +32, +64, +96  (4 full-EXEC issues)
#pragma unroll
        for (int j = 0; j < 4; ++j) {
            const int rr = prow + j * 32;
            async_lds_b64(lds_off(&wsb[b][rr * 18 + pc]), W,
                          (unsigned)(((nBase + rr) * K + k0 + pc) * 4));
        }
    };

    const int r  = lane & 15;           // row/col within fragment
    const int hk = (lane >> 4) << 1;    // k-pair select: 0 or 2

    issue(0, 0);
    for (int kc = 0; kc < 48; ++kc) {
        const int cur = kc & 1;
        if (kc < 47) {
            issue((kc + 1) * 16, 1 - cur);
            wait_async_6();             // oldest 6 (= chunk kc) complete
        } else {
            wait_async_0();
        }
        __syncthreads();

        const float* as_ = &asb[cur][0];
        const float* ws_ = &wsb[cur][0];
#pragma unroll
        for (int s = 0; s < 4; ++s) {
            const int kb = s * 4 + hk;
            v2f af = { as_[(mt * 16 + r) * 18 + kb],
                       as_[(mt * 16 + r) * 18 + kb + 1] };
#pragma unroll
            for (int t = 0; t < 4; ++t) {
                v2f bf = { ws_[((ng + t) * 16 + r) * 18 + kb],
                           ws_[((ng + t) * 16 + r) * 18 + kb + 1] };
                acc[t] = wmma4(af, bf, acc[t]);
            }
        }
        __syncthreads();                // buffer reused by issue() next iter
    }

    // C/D layout: vgpr i -> row i (lanes 0-15) / row 8+i (lanes 16-31), col = lane&15
    const int rbase = mBase + mt * 16 + ((lane >> 4) << 3);
#pragma unroll
    for (int t = 0; t < 4; ++t) {
        const int col = nBase + (ng + t) * 16 + r;
        const float bv = bias[col];
#pragma unroll
        for (int i = 0; i < 8; ++i)
            C[(size_t)(rbase + i) * N + col] = acc[t][i] + bv;
    }
}

// ---------------------------------------------------------------------------
// Attention per head slab (2048 x 48). Head (b,h) lives contiguously at
// offset (b*16+h)*2048*48 (the "raw view" reshape quirk). Scores are
// (q.k) * sqrt(48) (the "/scale" quirk) -- folded into Q on load.
// Block: 256 threads = 8 waves; wave owns 16 query rows; 128 q-rows/block.
// Flash-style online softmax; K/V chunks double-buffered via async->LDS DMA.
// ---------------------------------------------------------------------------
__global__ __launch_bounds__(256) void attn_kernel(
    const float* __restrict__ qbuf, const float* __restrict__ kbuf,
    const float* __restrict__ vbuf, float* __restrict__ obuf) {
    const int HD = 48;
    __shared__ float qs[128 * 49];        // 128 x 48, stride 49 (pre-scaled)
    __shared__ float ks[2][16 * 49];      // double-buffered 16 x 48
    __shared__ float vs[2][16 * 49];
    __shared__ float ps[8][16 * 17];      // per-wave P scratch (16x16)

    const int tid  = threadIdx.x;
    const int lane = tid & 31;
    const int wave = tid >> 5;
    const int bh   = blockIdx.y;                    // b*16 + h
    const int qBase = blockIdx.x * 128;
    const size_t slab = (size_t)bh * 2048 * HD;
    const float* qh = qbuf + slab;
    const float* kh = kbuf + slab;
    const float* vh = vbuf + slab;

    // K/V chunk staging: 768 floats each -> 3 full-EXEC B32 issues per array
    auto issueKV = [&](int keyBase, int b) {
#pragma unroll
        for (int j = 0; j < 3; ++j) {
            const int idx = j * 256 + tid;                 // 0..767
            const int rr = idx / HD, cc = idx - rr * HD;
            const unsigned gb = (unsigned)(((keyBase + rr) * HD + cc) * 4);
            async_lds_b32(lds_off(&ks[b][rr * 49 + cc]), kh, gb);
            async_lds_b32(lds_off(&vs[b][rr * 49 + cc]), vh, gb);
        }
    };

    issueKV(0, 0);

    const float SCALE = 6.92820323027551f;          // sqrt(48)
    for (int idx = tid; idx < 128 * HD; idx += 256) {
        int rr = idx / HD, cc = idx - rr * HD;
        qs[rr * 49 + cc] = qh[(size_t)(qBase + rr) * HD + cc] * SCALE;
    }

    const v8f vzero = {0.f,0.f,0.f,0.f,0.f,0.f,0.f,0.f};
    v8f o0 = vzero, o1 = vzero, o2 = vzero;
    float mrow[8], lrow[8];
#pragma unroll
    for (int i = 0; i < 8; ++i) { mrow[i] = -3.0e38f; lrow[i] = 0.f; }

    const int r  = lane & 15;
    const int hk = (lane >> 4) << 1;

    for (int kc = 0; kc < 128; ++kc) {
        const int cur = kc & 1;
        if (kc < 127) {
            issueKV((kc + 1) * 16, 1 - cur);
            wait_async_6();             // oldest 6 (= chunk kc) complete
        } else {
            wait_async_0();
        }
        __syncthreads();

        // S = Qtile (16x48) x K^T (48x16): B[k][n] = K[n][k] -> ks[n*49+k]
        const float* ksc = &ks[cur][0];
        const float* vsc = &vs[cur][0];
        v8f s = vzero;
#pragma unroll
        for (int sc = 0; sc < 12; ++sc) {
            const int kb = sc * 4 + hk;
            v2f af = { qs[(wave * 16 + r) * 49 + kb],
                       qs[(wave * 16 + r) * 49 + kb + 1] };
            v2f bf = { ksc[r * 49 + kb], ksc[r * 49 + kb + 1] };
            s = wmma4(af, bf, s);
        }

        // online softmax: stats per (vgpr row, half-wave); reduce across 16 lanes
        float alpha[8];
#pragma unroll
        for (int i = 0; i < 8; ++i) {
            float v = s[i];
#pragma unroll
            for (int m = 1; m < 16; m <<= 1)
                v = fmaxf(v, __shfl_xor(v, m, 32));
            const float mnew = fmaxf(mrow[i], v);
            alpha[i] = __expf(mrow[i] - mnew);
            mrow[i] = mnew;
            const float p = __expf(s[i] - mnew);
            s[i] = p;
            float rs = p;
#pragma unroll
            for (int m = 1; m < 16; m <<= 1)
                rs += __shfl_xor(rs, m, 32);
            lrow[i] = lrow[i] * alpha[i] + rs;
            o0[i] *= alpha[i]; o1[i] *= alpha[i]; o2[i] *= alpha[i];
        }

        // C-layout P -> LDS -> A-layout fragments (within-wave, no barrier)
        float* pw = &ps[wave][0];
        const int rowoff = (lane >> 4) << 3;
#pragma unroll
        for (int i = 0; i < 8; ++i)
            pw[(rowoff + i) * 17 + r] = s[i];

        // O (16x48) += P (16x16) x Vchunk (16x48)
#pragma unroll
        for (int sc = 0; sc < 4; ++sc) {
            const int kb = sc * 4 + hk;
            v2f af = { pw[r * 17 + kb], pw[r * 17 + kb + 1] };
            v2f b0 = { vsc[kb * 49 + r],      vsc[(kb + 1) * 49 + r] };
            v2f b1 = { vsc[kb * 49 + 16 + r], vsc[(kb + 1) * 49 + 16 + r] };
            v2f b2 = { vsc[kb * 49 + 32 + r], vsc[(kb + 1) * 49 + 32 + r] };
            o0 = wmma4(af, b0, o0);
            o1 = wmma4(af, b1, o1);
            o2 = wmma4(af, b2, o2);
        }
        __syncthreads();                // buffer reused by issueKV next iter
    }

    float* oh = obuf + slab;
    const int rbase = qBase + wave * 16 + ((lane >> 4) << 3);
#pragma unroll
    for (int i = 0; i < 8; ++i) {
        const float inv = 1.0f / lrow[i];
        oh[(size_t)(rbase + i) * HD + r]      = o0[i] * inv;
        oh[(size_t)(rbase + i) * HD + 16 + r] = o1[i] * inv;
        oh[(size_t)(rbase + i) * HD + 32 + r] = o2[i] * inv;
    }
}

// ---------------------------------------------------------------------------
extern "C" void kernel_launch(void* const* d_in, const int* in_sizes, int n_in,
                              void* d_out, int out_size, void* d_ws, size_t ws_size,
                              hipStream_t stream) {
    (void)in_sizes; (void)n_in; (void)out_size; (void)ws_size;
    const float* x  = (const float*)d_in[0];
    const float* Wq = (const float*)d_in[1];
    const float* bq = (const float*)d_in[2];
    const float* Wk = (const float*)d_in[3];
    const float* bk = (const float*)d_in[4];
    const float* Wv = (const float*)d_in[5];
    const float* bv = (const float*)d_in[6];
    const float* Wo = (const float*)d_in[7];
    const float* bo = (const float*)d_in[8];
    float* out = (float*)d_out;
    float* ws  = (float*)d_ws;

    const size_t SEG = (size_t)8192 * 768;   // one (B*N, D) fp32 buffer
    float* qb = ws;
    float* kb = ws + SEG;
    float* vb = ws + 2 * SEG;
    float* ab = ws + 3 * SEG;

    dim3 gemmGrid(128, 6);       // 8192/64 x 768/128
    dim3 attnGrid(16, 64);       // 2048/128 q-chunks x (B*H) heads

    gemm_xwT_bias<<<gemmGrid, 256, 0, stream>>>(x, Wq, bq, qb);
    gemm_xwT_bias<<<gemmGrid, 256, 0, stream>>>(x, Wk, bk, kb);
    gemm_xwT_bias<<<gemmGrid, 256, 0, stream>>>(x, Wv, bv, vb);
    attn_kernel<<<attnGrid, 256, 0, stream>>>(qb, kb, vb, ab);
    gemm_xwT_bias<<<gemmGrid, 256, 0, stream>>>(ab, Wo, bo, out);
}